// DifferentiableEKF_1305670058698
// MI455X (gfx1250) — compile-verified
//
#include <hip/hip_runtime.h>
#include <math.h>

// Problem constants (match reference)
#define BB 1024
#define TT 200
#define M_ 16
#define N_ 8
#define H_ 64
#define DT_ 0.01f
#define JIT_ 1e-6f

typedef __attribute__((ext_vector_type(2))) float v2f;
typedef __attribute__((ext_vector_type(8))) float v8f;

// ---- WMMA helpers: D(16x16) = A(16x16) @ B(16x16) (+C), f32, via 4 chained
// V_WMMA_F32_16X16X4_F32. Fragment layouts per CDNA5 ISA 7.12.2:
//   A 16x4: lane<16 -> row=lane,   v0:K=k0,  v1:K=k0+1
//           lane>=16-> row=lane-16,v0:K=k0+2,v1:K=k0+3
//   B 4x16: same K split across lane halves, N = lane&15
//   C/D:    VGPR v holds row (v + 8*(lane>>4)), col = lane&15
__device__ __forceinline__ v8f wmma_mm16(const float* A, const float* B,
                                         bool transB, v8f acc, int lane) {
  const int half = lane >> 4;
  const int r = lane & 15;
#pragma unroll
  for (int kk = 0; kk < 4; ++kk) {
    const int k0 = 4 * kk + 2 * half;
    v2f a, bb;
    a[0] = A[r * 16 + k0];
    a[1] = A[r * 16 + k0 + 1];
    if (!transB) {              // B[k][n]
      bb[0] = B[k0 * 16 + r];
      bb[1] = B[(k0 + 1) * 16 + r];
    } else {                    // B = X^T  => B[k][n] = X[n][k]
      bb[0] = B[r * 16 + k0];
      bb[1] = B[r * 16 + k0 + 1];
    }
    acc = __builtin_amdgcn_wmma_f32_16x16x4_f32(false, a, false, bb,
                                                (short)0, acc, false, false);
  }
  return acc;
}

__device__ __forceinline__ v8f load_frag(const float* Cm, int lane) {
  const int half = lane >> 4, r = lane & 15;
  v8f c;
#pragma unroll
  for (int v = 0; v < 8; ++v) c[v] = Cm[(v + 8 * half) * 16 + r];
  return c;
}

__device__ __forceinline__ void store_frag(float* Dm, v8f d, int lane) {
  const int half = lane >> 4, r = lane & 15;
#pragma unroll
  for (int v = 0; v < 8; ++v) Dm[(v + 8 * half) * 16 + r] = d[v];
}

__global__ __launch_bounds__(32) void ekf_scan_kernel(
    const float* __restrict__ gObs, const float* __restrict__ gX0,
    const float* __restrict__ gS0, const float* __restrict__ gW1,
    const float* __restrict__ gB1, const float* __restrict__ gW2,
    const float* __restrict__ gB2, const float* __restrict__ gC,
    const float* __restrict__ gD, const float* __restrict__ gLq,
    const float* __restrict__ gLr,
    float* __restrict__ oXp, float* __restrict__ oXn,
    float* __restrict__ oSp, float* __restrict__ oSn,
    float* __restrict__ oDy, float* __restrict__ oS,
    float* __restrict__ oLd, float* __restrict__ oW) {
  __shared__ float sW1[M_ * H_];   // W1 row-major (16x64)
  __shared__ float sW2[H_ * M_];   // W2 row-major (64x16)
  __shared__ float sX[M_], sXp[M_], sH[H_], sG[H_];
  __shared__ float sSg[256];       // Sigma (state)
  __shared__ float sSp[256];       // Sigma_pred
  __shared__ float sTm[256];       // temp
  __shared__ float sSn[256];       // Sigma_new accum
  __shared__ float sFm[256];       // F, then ImKH
  __shared__ float sKp[256];       // K padded (cols 8..15 == 0)
  __shared__ float sQ[256];        // Q
  __shared__ float sRp[256];       // R padded 16x16
  __shared__ float sCp[256];       // C padded 16x16 (rows 8..15 == 0)
  __shared__ float sS[64], sL[64], sDy[8], sU[128];  // U = Sigma_pred C^T (16x8)

  const int lane = threadIdx.x;
  const int b = blockIdx.x;

  // ---------- one-time init per wave ----------
#pragma unroll 4
  for (int r = 0; r < 32; ++r) sW1[lane + 32 * r] = gW1[lane + 32 * r];
#pragma unroll 4
  for (int r = 0; r < 32; ++r) sW2[lane + 32 * r] = gW2[lane + 32 * r];
  if (lane < 16) sX[lane] = gX0[b * 16 + lane];
#pragma unroll 1
  for (int r = 0; r < 8; ++r) {
    const int e = lane + 32 * r;
    sSg[e] = gS0[b * 256 + e];
    const int i = e >> 4, j = e & 15;
    float q = (i == j) ? 1e-4f : 0.0f;
    for (int k = 0; k < 16; ++k) q += gLq[i * 16 + k] * gLq[j * 16 + k];
    sQ[e] = q;
    float rp = 0.0f;
    if (i < 8 && j < 8) {
      rp = (i == j) ? 1e-4f : 0.0f;
      for (int k = 0; k < 8; ++k) rp += gLr[i * 8 + k] * gLr[j * 8 + k];
    }
    sRp[e] = rp;
    sCp[e] = (i < 8) ? gC[i * 16 + j] : 0.0f;
    sKp[e] = 0.0f;
  }
  __syncthreads();

  const v8f vz = {0, 0, 0, 0, 0, 0, 0, 0};

  // ---------- sequential scan over time ----------
#pragma unroll 1
  for (int t = 0; t < TT; ++t) {
    const size_t bt = (size_t)b * TT + t;
    if (t + 1 < TT) __builtin_prefetch(gObs + bt * 8 + 8, 0, 0);

    // (1) h = tanh(x@W1 + b1), g = 1 - h^2
#pragma unroll
    for (int kb = 0; kb < 2; ++kb) {
      const int k = lane + 32 * kb;
      float acc = gB1[k];
      for (int i = 0; i < 16; ++i) acc += sX[i] * sW1[i * 64 + k];
      const float hv = tanhf(acc);
      sH[k] = hv;
      sG[k] = 1.0f - hv * hv;
    }
    __syncthreads();

    // (2) x_pred = x + dt*(h@W2 + b2)
    if (lane < 16) {
      float acc = gB2[lane];
      for (int k = 0; k < 64; ++k) acc += sH[k] * sW2[k * 16 + lane];
      const float xp = sX[lane] + DT_ * acc;
      sXp[lane] = xp;
      oXp[bt * 16 + lane] = xp;
    }
    __syncthreads();

    // (3) F = I + dt * W2^T diag(g) W1^T   (16x16x64 WMMA chain)
    {
      const int half = lane >> 4, r = lane & 15;
      v8f acc = vz;
#pragma unroll
      for (int kk = 0; kk < 16; ++kk) {
        const int k0 = 4 * kk + 2 * half;
        v2f a, bb;
        a[0] = sW2[k0 * 16 + r];          // A[j][k] = W2[k][j], row j=r
        a[1] = sW2[(k0 + 1) * 16 + r];
        bb[0] = sG[k0] * sW1[r * 64 + k0];        // B[k][i] = g[k]*W1[i][k]
        bb[1] = sG[k0 + 1] * sW1[r * 64 + k0 + 1];
        acc = __builtin_amdgcn_wmma_f32_16x16x4_f32(false, a, false, bb,
                                                    (short)0, acc, false, false);
      }
#pragma unroll
      for (int v = 0; v < 8; ++v) {
        const int row = v + 8 * half;
        sFm[row * 16 + r] = ((row == r) ? 1.0f : 0.0f) + DT_ * acc[v];
      }
    }
    __syncthreads();

    // (4) Sigma_pred = F Sigma F^T + Q
    {
      v8f tf = wmma_mm16(sFm, sSg, false, vz, lane);
      store_frag(sTm, tf, lane);
      __syncthreads();
      v8f cq = load_frag(sQ, lane);
      v8f sp = wmma_mm16(sTm, sFm, true, cq, lane);
      store_frag(sSp, sp, lane);
      const int half = lane >> 4, r = lane & 15;
#pragma unroll
      for (int v = 0; v < 8; ++v)
        oSp[bt * 256 + (v + 8 * half) * 16 + r] = sp[v];
    }
    __syncthreads();

    // (5) U = Sigma_pred @ C^T   (16x8)
#pragma unroll
    for (int rr = 0; rr < 4; ++rr) {
      const int e = lane + 32 * rr;      // e == i*8 + n
      const int i = e >> 3, n = e & 7;
      float acc = 0.0f;
      for (int j = 0; j < 16; ++j) acc += sSp[i * 16 + j] * sCp[n * 16 + j];
      sU[e] = acc;
    }
    __syncthreads();

    // (6) S = C U + R  (output pre-jitter), then sL = S + jitter*I
#pragma unroll
    for (int rr = 0; rr < 2; ++rr) {
      const int e = lane + 32 * rr;      // e == n*8 + l
      const int n = e >> 3, l = e & 7;
      float acc = sRp[n * 16 + l];
      for (int i = 0; i < 16; ++i) acc += sCp[n * 16 + i] * sU[i * 8 + l];
      sS[e] = acc;
      oS[bt * 64 + e] = acc;
      sL[e] = acc + ((n == l) ? JIT_ : 0.0f);
    }
    __syncthreads();

    // (7) delta_y = obs - (x_pred @ C^T + d)
    if (lane < 8) {
      float yp = gD[lane];
      for (int i = 0; i < 16; ++i) yp += sXp[i] * sCp[lane * 16 + i];
      const float dv = gObs[bt * 8 + lane] - yp;
      sDy[lane] = dv;
      oDy[bt * 8 + lane] = dv;
    }
    __syncthreads();

    // (8) in-place lower Cholesky of 8x8 in sL (lane-parallel per column)
    for (int k = 0; k < 8; ++k) {
      if (lane == k) {
        float dk = sL[k * 8 + k];
        for (int p = 0; p < k; ++p) dk -= sL[k * 8 + p] * sL[k * 8 + p];
        sL[k * 8 + k] = sqrtf(fmaxf(dk, 1e-30f));
      }
      __syncthreads();
      if (lane > k && lane < 8) {
        float v = sL[lane * 8 + k];
        for (int p = 0; p < k; ++p) v -= sL[lane * 8 + p] * sL[k * 8 + p];
        sL[lane * 8 + k] = v / sL[k * 8 + k];
      }
      __syncthreads();
    }

    // (9) logdet + whitened = L^{-1} dy   (lane 0)
    if (lane == 0) {
      float acc = 0.0f, w[8];
      for (int n = 0; n < 8; ++n) {
        acc += logf(sL[n * 8 + n]);
        float v = sDy[n];
        for (int p = 0; p < n; ++p) v -= sL[n * 8 + p] * w[p];
        w[n] = v / sL[n * 8 + n];
        oW[bt * 8 + n] = w[n];
      }
      oLd[bt] = 2.0f * acc;
    }

    // (10) K row j: solve (L L^T) z = U[j,:]   (16 lanes, one row each)
    if (lane < 16) {
      float v[8];
      for (int n = 0; n < 8; ++n) v[n] = sU[lane * 8 + n];
      for (int n = 0; n < 8; ++n) {          // forward
        float x = v[n];
        for (int p = 0; p < n; ++p) x -= sL[n * 8 + p] * v[p];
        v[n] = x / sL[n * 8 + n];
      }
      for (int n = 7; n >= 0; --n) {         // backward (L^T)
        float x = v[n];
        for (int p = n + 1; p < 8; ++p) x -= sL[p * 8 + n] * v[p];
        v[n] = x / sL[n * 8 + n];
      }
      for (int n = 0; n < 8; ++n) sKp[lane * 16 + n] = v[n];  // cols 8..15 stay 0
    }
    __syncthreads();

    // (11) x_new = x_pred + K dy  (becomes next state)
    if (lane < 16) {
      float xn = sXp[lane];
      for (int n = 0; n < 8; ++n) xn += sKp[lane * 16 + n] * sDy[n];
      sX[lane] = xn;
      oXn[bt * 16 + lane] = xn;
    }
    __syncthreads();

    // (12) ImKH = I - K C  (padded 16x16 WMMA)
    {
      v8f kc = wmma_mm16(sKp, sCp, false, vz, lane);
      const int half = lane >> 4, r = lane & 15;
#pragma unroll
      for (int v = 0; v < 8; ++v) {
        const int row = v + 8 * half;
        sFm[row * 16 + r] = ((row == r) ? 1.0f : 0.0f) - kc[v];
      }
    }
    __syncthreads();

    // (13) Sa = ImKH Sigma_pred ImKH^T
    {
      v8f t1 = wmma_mm16(sFm, sSp, false, vz, lane);
      store_frag(sTm, t1, lane);
      __syncthreads();
      v8f sa = wmma_mm16(sTm, sFm, true, vz, lane);
      store_frag(sSn, sa, lane);
    }
    __syncthreads();

    // (14) Sigma_new = Sa + (K R) K^T
    {
      v8f kr = wmma_mm16(sKp, sRp, false, vz, lane);
      store_frag(sTm, kr, lane);
      __syncthreads();
      v8f cs = load_frag(sSn, lane);
      v8f sn = wmma_mm16(sTm, sKp, true, cs, lane);
      store_frag(sSn, sn, lane);
    }
    __syncthreads();

    // (15) symmetrize -> state + output
#pragma unroll
    for (int rr = 0; rr < 8; ++rr) {
      const int e = lane + 32 * rr;
      const int i = e >> 4, j = e & 15;
      const float val = 0.5f * (sSn[i * 16 + j] + sSn[j * 16 + i]);
      sSg[e] = val;
      oSn[bt * 256 + e] = val;
    }
    __syncthreads();
  }
}

extern "C" void kernel_launch(void* const* d_in, const int* in_sizes, int n_in,
                              void* d_out, int out_size, void* d_ws,
                              size_t ws_size, hipStream_t stream) {
  (void)in_sizes; (void)n_in; (void)d_ws; (void)ws_size; (void)out_size;
  const float* gObs = (const float*)d_in[0];
  const float* gX0  = (const float*)d_in[1];
  const float* gS0  = (const float*)d_in[2];
  const float* gW1  = (const float*)d_in[3];
  const float* gB1  = (const float*)d_in[4];
  const float* gW2  = (const float*)d_in[5];
  const float* gB2  = (const float*)d_in[6];
  const float* gC   = (const float*)d_in[7];
  const float* gD   = (const float*)d_in[8];
  const float* gLq  = (const float*)d_in[9];
  const float* gLr  = (const float*)d_in[10];

  float* out = (float*)d_out;
  const size_t BT = (size_t)BB * TT;
  float* oXp = out;                 // (B,T,16)
  float* oXn = out + BT * 16;       // (B,T,16)
  float* oSp = out + BT * 32;       // (B,T,16,16)
  float* oSn = out + BT * 288;      // (B,T,16,16)
  float* oDy = out + BT * 544;      // (B,T,8)
  float* oS  = out + BT * 552;      // (B,T,8,8)
  float* oLd = out + BT * 616;      // (B,T)
  float* oW  = out + BT * 617;      // (B,T,8)

  hipLaunchKernelGGL(ekf_scan_kernel, dim3(BB), dim3(32), 0, stream,
                     gObs, gX0, gS0, gW1, gB1, gW2, gB2, gC, gD, gLq, gLr,
                     oXp, oXn, oSp, oSn, oDy, oS, oLd, oW);
}